// LayeredShader_86517821214071
// MI455X (gfx1250) — compile-verified
//
#include <hip/hip_runtime.h>
#include <math.h>

// ---------------- problem constants (from reference) ----------------
#define SIGMA_F      1e-4f
#define N_IMG        8
#define H_IMG        512
#define W_IMG        512
#define K_LAYERS     8
#define PIX_PER_BLK  256          // 8 wave32s per block
#define HW_PLANE     (H_IMG * W_IMG)

// exp(-x/sigma) = exp2(x * (-log2(e)/sigma))
#define EXP_SCALE    (-14426.950408889634f)

// ---------------- gfx1250 feature probes ----------------
#if defined(__gfx1250__) && \
    __has_builtin(__builtin_amdgcn_global_load_async_to_lds_b128) && \
    __has_builtin(__builtin_amdgcn_s_wait_asynccnt)
#define USE_ASYNC_LDS 1
#else
#define USE_ASYNC_LDS 0
#endif

// Native (clang ext) vector types -- required by the async-LDS builtin's
// parameter types and by __builtin_nontemporal_load.
typedef int   v4i __attribute__((ext_vector_type(4)));
typedef float v4f __attribute__((ext_vector_type(4)));
typedef __attribute__((address_space(1))) v4i* gv4i_p;   // global
typedef __attribute__((address_space(3))) v4i* lv4i_p;   // LDS

__device__ __forceinline__ float fast_exp2(float x) {
#if __has_builtin(__builtin_amdgcn_exp2f)
    return __builtin_amdgcn_exp2f(x);     // v_exp_f32
#else
    return exp2f(x);
#endif
}

// One block = 256 pixels. Stage the block's slab of colors/dists/faces into
// LDS with lane-contiguous b128 async copies, then composite per-thread.
__global__ __launch_bounds__(PIX_PER_BLK) void
layered_shader_kernel(const float* __restrict__ colors,     // (N,H,W,K,3)
                      const float* __restrict__ dists,      // (N,H,W,K)
                      const float* __restrict__ background, // (3,)
                      const int*   __restrict__ p2f,        // (N,H,W,K) int32
                      float* __restrict__ out)              // (N,4,H,W)
{
    __shared__ __align__(16) float s_col[PIX_PER_BLK * 24];  // 24 KB
    __shared__ __align__(16) float s_dst[PIX_PER_BLK * 8];   //  8 KB
    __shared__ __align__(16) int   s_p2f[PIX_PER_BLK * 8];   //  8 KB

    const int tid     = threadIdx.x;
    const int pixBase = blockIdx.x * PIX_PER_BLK;

    const char* gcol = (const char*)(colors + (long long)pixBase * 24);
    const char* gdst = (const char*)(dists  + (long long)pixBase * 8);
    const char* gpf  = (const char*)(p2f    + (long long)pixBase * 8);

#if USE_ASYNC_LDS
    // colors: 24576 B = 1536 x 16B chunks -> 6 per thread
    #pragma unroll
    for (int i = 0; i < 6; ++i) {
        const int c = tid + i * PIX_PER_BLK;
        __builtin_amdgcn_global_load_async_to_lds_b128(
            (gv4i_p)(gcol + c * 16), (lv4i_p)((char*)s_col + c * 16), 0, 0);
    }
    // dists / p2f: 8192 B each = 512 chunks -> 2 per thread each
    #pragma unroll
    for (int i = 0; i < 2; ++i) {
        const int c = tid + i * PIX_PER_BLK;
        __builtin_amdgcn_global_load_async_to_lds_b128(
            (gv4i_p)(gdst + c * 16), (lv4i_p)((char*)s_dst + c * 16), 0, 0);
        __builtin_amdgcn_global_load_async_to_lds_b128(
            (gv4i_p)(gpf  + c * 16), (lv4i_p)((char*)s_p2f + c * 16), 0, 0);
    }
    __builtin_amdgcn_s_wait_asynccnt(0);
#else
    // Fallback: identical lane-contiguous staging through VGPRs.
    {
        const v4f* g4 = (const v4f*)gcol;
        v4f*       s4 = (v4f*)s_col;
        #pragma unroll
        for (int i = 0; i < 6; ++i) {
            const int c = tid + i * PIX_PER_BLK;
            s4[c] = __builtin_nontemporal_load(&g4[c]);
        }
        const v4f* g4d = (const v4f*)gdst;
        v4f*       s4d = (v4f*)s_dst;
        const v4i* g4p = (const v4i*)gpf;
        v4i*       s4p = (v4i*)s_p2f;
        #pragma unroll
        for (int i = 0; i < 2; ++i) {
            const int c = tid + i * PIX_PER_BLK;
            s4d[c] = __builtin_nontemporal_load(&g4d[c]);
            s4p[c] = __builtin_nontemporal_load(&g4p[c]);
        }
    }
#endif
    __syncthreads();

    const float b0 = background[0];
    const float b1 = background[1];
    const float b2 = background[2];

    const float* col = s_col + tid * 24;
    const float* dd  = s_dst + tid * 8;
    const int*   pf  = s_p2f + tid * 8;

    float occ = 1.0f;                 // running prod(1 - alpha)
    float r0 = 0.0f, r1 = 0.0f, r2 = 0.0f;

    #pragma unroll
    for (int k = 0; k < K_LAYERS; ++k) {
        const float t  = fmaxf(dd[k], 0.0f) * EXP_SCALE;   // <= 0
        float alpha    = fast_exp2(t);                     // v_exp_f32
        alpha          = (pf[k] >= 0) ? alpha : 0.0f;      // face mask
        const float w  = occ * alpha;                      // occ BEFORE layer k
        r0 = fmaf(w, col[k * 3 + 0], r0);
        r1 = fmaf(w, col[k * 3 + 1], r1);
        r2 = fmaf(w, col[k * 3 + 2], r2);
        occ *= (1.0f - alpha);
    }
    // background term: occ_final * 1 * bg
    r0 = fmaf(occ, b0, r0);
    r1 = fmaf(occ, b1, r1);
    r2 = fmaf(occ, b2, r2);
    const float a = 1.0f - occ;

    // output (N,4,H,W): coalesced per-plane stores, streamed -> non-temporal
    const int pix = pixBase + tid;
    const int n   = pix / HW_PLANE;
    const int r   = pix - n * HW_PLANE;
    float* o = out + (long long)n * 4 * HW_PLANE + r;
    __builtin_nontemporal_store(r0, o + 0LL * HW_PLANE);
    __builtin_nontemporal_store(r1, o + 1LL * HW_PLANE);
    __builtin_nontemporal_store(r2, o + 2LL * HW_PLANE);
    __builtin_nontemporal_store(a,  o + 3LL * HW_PLANE);
}

extern "C" void kernel_launch(void* const* d_in, const int* in_sizes, int n_in,
                              void* d_out, int out_size, void* d_ws, size_t ws_size,
                              hipStream_t stream) {
    (void)in_sizes; (void)n_in; (void)out_size; (void)d_ws; (void)ws_size;
    const float* colors     = (const float*)d_in[0];   // (N,H,W,K,3) f32
    const float* dists      = (const float*)d_in[1];   // (N,H,W,K)   f32
    const float* background = (const float*)d_in[2];   // (3,)        f32
    const int*   p2f        = (const int*)d_in[3];     // (N,H,W,K)   int32
    float*       out        = (float*)d_out;           // (N,4,H,W)   f32

    const int totalPix = N_IMG * H_IMG * W_IMG;        // 2,097,152
    const int blocks   = totalPix / PIX_PER_BLK;       // 8192, exact
    layered_shader_kernel<<<blocks, PIX_PER_BLK, 0, stream>>>(
        colors, dists, background, p2f, out);
}